// GCNSampling_37967510896973
// MI455X (gfx1250) — compile-verified
//
#include <hip/hip_runtime.h>

typedef float v2f __attribute__((ext_vector_type(2)));
typedef float v4f __attribute__((ext_vector_type(4)));
typedef float v8f __attribute__((ext_vector_type(8)));

#define IN    512
#define HID   128
#define HID2  256     // 2*HID
#define NCLS  64
#define KNB   10
#define N1    40960
#define N2    4096

#define LDX1  (IN  + 4)   // LDS row stride (floats): 516 % 64 = 4 -> conflict-free A reads
#define LDX2  (HID2 + 4)  // 260

// ---------------------------------------------------------------------------
// Kernel 1: fused [gather-mean over nbr0 of (embed - h0_hist)] + agg_h0
//           -> x1 tile (16 x 512) in LDS
//           -> x1 @ W1 + b1 via v_wmma_f32_16x16x4_f32 (8 waves, N=128)
//           -> z1 = concat(y, relu(y)) - h1_hist   (written to workspace)
// ---------------------------------------------------------------------------
__global__ __launch_bounds__(256) void
gcn_k1(const float* __restrict__ embed,
       const float* __restrict__ h0_hist,
       const float* __restrict__ h1_hist,
       const float* __restrict__ agg_h0,
       const float* __restrict__ W1,
       const float* __restrict__ b1,
       const int*   __restrict__ nbr0,
       float*       __restrict__ z1)
{
    __shared__ float X[16 * LDX1];     // 33,024 B
    __shared__ int   snbr[16 * KNB];   // neighbor indices for this tile

    const int tid  = threadIdx.x;
    const int tile = blockIdx.x;       // 16 dst rows per workgroup

    if (tid < 16 * KNB)
        snbr[tid] = nbr0[tile * 16 * KNB + tid];
    __syncthreads();

    // ---- gather + mean: 2 row-groups x 128 threads, float4 (b128) loads ----
    const int g  = tid >> 7;           // row group 0/1
    const int t  = tid & 127;
    const int c0 = t * 4;              // 128 threads x 4 floats = 512 cols

    for (int rr = 0; rr < 8; ++rr) {
        const int r = rr * 2 + g;
        v4f s = {0.f, 0.f, 0.f, 0.f};
        #pragma unroll
        for (int k = 0; k < KNB; ++k) {
            const int src = snbr[r * KNB + k];
            // single-touch streams: non-temporal, don't pollute L2
            const v4f e = __builtin_nontemporal_load(
                (const v4f*)(embed   + (size_t)src * IN + c0));
            const v4f h = __builtin_nontemporal_load(
                (const v4f*)(h0_hist + (size_t)src * IN + c0));
            s += e - h;
        }
        const v4f ah = *(const v4f*)(agg_h0 + (size_t)(tile * 16 + r) * IN + c0);
        *(v4f*)(&X[r * LDX1 + c0]) = s * (1.f / KNB) + ah;  // ds_store_b128
    }
    __syncthreads();

    // ---- GEMM: X[16,512] @ W1[512,128] ; each wave owns one 16x16 N-tile ----
    const int wave = tid >> 5;         // 0..7  -> N block
    const int lane = tid & 31;
    const int half = lane >> 4;        // A/B: K split across lane halves
    const int lm   = lane & 15;        // row (A) / col (B)
    const int n0   = wave * 16;

    v8f acc0 = {}, acc1 = {};          // two chains -> more XDL ILP
    #pragma unroll 2
    for (int k0 = 0; k0 < IN; k0 += 8) {
        // A (16x4 f32): V0 = {K | K+2}, V1 = {K+1 | K+3}; even idx -> ds_load_b64
        v2f a0 = *(const v2f*)(&X[lm * LDX1 + k0 + 2 * half]);
        v2f a1 = *(const v2f*)(&X[lm * LDX1 + k0 + 4 + 2 * half]);
        const float* wp0 = W1 + (size_t)(k0 + 2 * half) * HID + n0 + lm;
        const float* wp1 = wp0 + (size_t)4 * HID;
        v2f b0, b1v;
        b0.x  = wp0[0];  b0.y  = wp0[HID];
        b1v.x = wp1[0];  b1v.y = wp1[HID];
        acc0 = __builtin_amdgcn_wmma_f32_16x16x4_f32(
                   false, a0, false, b0,  (short)0, acc0, false, false);
        acc1 = __builtin_amdgcn_wmma_f32_16x16x4_f32(
                   false, a1, false, b1v, (short)0, acc1, false, false);
    }

    // ---- epilogue: bias, concat(y, relu(y)), subtract h1_hist, store z1 ----
    #pragma unroll
    for (int j = 0; j < 8; ++j) {
        const int m   = j + 8 * half;  // C/D layout: VGPR j = rows j / j+8
        const int n   = n0 + lm;
        const int row = tile * 16 + m;
        const float y  = acc0[j] + acc1[j] + b1[n];
        const float yr = y > 0.f ? y : 0.f;
        const size_t base = (size_t)row * HID2;
        z1[base + n]       = y  - h1_hist[base + n];
        z1[base + HID + n] = yr - h1_hist[base + HID + n];
    }
}

// ---------------------------------------------------------------------------
// Kernel 2: fused [gather-mean over nbr1 of z1] + agg_h1
//           -> x2 tile (16 x 256) in LDS
//           -> x2 @ W2 + b2 via v_wmma_f32_16x16x4_f32 (4 waves, N=64)
// ---------------------------------------------------------------------------
__global__ __launch_bounds__(128) void
gcn_k2(const float* __restrict__ z1,
       const float* __restrict__ agg_h1,
       const float* __restrict__ W2,
       const float* __restrict__ b2,
       const int*   __restrict__ nbr1,
       float*       __restrict__ out)
{
    __shared__ float X[16 * LDX2];     // 16,640 B
    __shared__ int   snbr[16 * KNB];

    const int tid  = threadIdx.x;
    const int tile = blockIdx.x;

    for (int i = tid; i < 16 * KNB; i += 128)
        snbr[i] = nbr1[tile * 16 * KNB + i];
    __syncthreads();

    // ---- gather + mean: 2 row-groups x 64 threads, float4 loads ----
    const int g  = tid >> 6;
    const int t  = tid & 63;
    const int c0 = t * 4;              // 64 threads x 4 floats = 256 cols

    for (int rr = 0; rr < 8; ++rr) {
        const int r = rr * 2 + g;
        v4f s = {0.f, 0.f, 0.f, 0.f};
        #pragma unroll
        for (int k = 0; k < KNB; ++k) {
            const int src = snbr[r * KNB + k];
            const v4f v = *(const v4f*)(z1 + (size_t)src * HID2 + c0);
            s += v;
        }
        const v4f ah = *(const v4f*)(agg_h1 + (size_t)(tile * 16 + r) * HID2 + c0);
        *(v4f*)(&X[r * LDX2 + c0]) = s * (1.f / KNB) + ah;
    }
    __syncthreads();

    const int wave = tid >> 5;         // 0..3
    const int lane = tid & 31;
    const int half = lane >> 4;
    const int lm   = lane & 15;
    const int n0   = wave * 16;

    v8f acc0 = {}, acc1 = {};
    #pragma unroll 2
    for (int k0 = 0; k0 < HID2; k0 += 8) {
        v2f a0 = *(const v2f*)(&X[lm * LDX2 + k0 + 2 * half]);
        v2f a1 = *(const v2f*)(&X[lm * LDX2 + k0 + 4 + 2 * half]);
        const float* wp0 = W2 + (size_t)(k0 + 2 * half) * NCLS + n0 + lm;
        const float* wp1 = wp0 + (size_t)4 * NCLS;
        v2f b0, b1v;
        b0.x  = wp0[0];  b0.y  = wp0[NCLS];
        b1v.x = wp1[0];  b1v.y = wp1[NCLS];
        acc0 = __builtin_amdgcn_wmma_f32_16x16x4_f32(
                   false, a0, false, b0,  (short)0, acc0, false, false);
        acc1 = __builtin_amdgcn_wmma_f32_16x16x4_f32(
                   false, a1, false, b1v, (short)0, acc1, false, false);
    }

    #pragma unroll
    for (int j = 0; j < 8; ++j) {
        const int m   = j + 8 * half;
        const int row = tile * 16 + m;
        const int n   = n0 + lm;
        out[(size_t)row * NCLS + n] = acc0[j] + acc1[j] + b2[n];
    }
}

// ---------------------------------------------------------------------------
extern "C" void kernel_launch(void* const* d_in, const int* in_sizes, int n_in,
                              void* d_out, int out_size, void* d_ws, size_t ws_size,
                              hipStream_t stream)
{
    const float* embed   = (const float*)d_in[0];
    const float* h0_hist = (const float*)d_in[1];
    const float* h1_hist = (const float*)d_in[2];
    const float* agg_h0  = (const float*)d_in[3];
    const float* agg_h1  = (const float*)d_in[4];
    const float* W1      = (const float*)d_in[5];
    const float* b1      = (const float*)d_in[6];
    const float* W2      = (const float*)d_in[7];
    const float* b2      = (const float*)d_in[8];
    const int*   nbr0    = (const int*)d_in[9];
    const int*   nbr1    = (const int*)d_in[10];

    float* z1 = (float*)d_ws;   // [N1, 256] fp32 = 41.9 MB scratch

    gcn_k1<<<N1 / 16, 256, 0, stream>>>(embed, h0_hist, h1_hist, agg_h0,
                                        W1, b1, nbr0, z1);
    gcn_k2<<<N2 / 16, 128, 0, stream>>>(z1, agg_h1, W2, b2, nbr1,
                                        (float*)d_out);
}